// DeformableTemporalAttention_76605036691589
// MI455X (gfx1250) — compile-verified
//
#include <hip/hip_runtime.h>
#include <hip/hip_bf16.h>
#include <math.h>

// ---------------------------------------------------------------------------
// DeformableTemporalAttention for MI455X (gfx1250, wave32, WMMA)
//   B=8, Q=2048, D=256, H=8, L=3, P=4, HD=32, HLP=96, Ts={2048,1024,512}
// ---------------------------------------------------------------------------

typedef float v2f __attribute__((ext_vector_type(2)));
typedef float v4f __attribute__((ext_vector_type(4)));
typedef float v8f __attribute__((ext_vector_type(8)));

#define D_MODEL 256
#define NT 32   // N tile per workgroup (LDS-staged weight columns)
#define MT 128  // M tile per workgroup (8 waves x 16 rows)

// ---------------------------------------------------------------------------
// Generic C[M,N] = A[M,256] @ W[256,N] + bias[N]   (f32, exact, WMMA 16x16x4)
//   grid = (M/128, N/32), block = 256 threads (8 waves)
//   Each wave: one 16x32 strip -> two 16x16 f32 accumulators.
//
// LDS layout is K-pair interleaved so a lane's B fragment (W[kb][n],W[kb+1][n])
// is a single aligned 8-byte ds_load_b64 directly into an even VGPR pair:
//   Wlds[(k>>1)*64 + n*2 + (k&1)] = W[k][n0+n]
// ---------------------------------------------------------------------------
__global__ __launch_bounds__(256)
void dta_gemm_f32_wmma(const float* __restrict__ A, const float* __restrict__ W,
                       const float* __restrict__ bias, float* __restrict__ C,
                       int M, int N) {
  __shared__ float Wlds[D_MODEL * NT]; // 32 KB, interleaved K-pairs

  const int tid = threadIdx.x;
  const int n0  = blockIdx.y * NT;

  // Cooperative staging: chunk = (pair-row kk, 4-column group).
  // Load b128 from rows 2kk and 2kk+1, store two shuffled b128 to LDS.
  // chunks = 128 * (32/4) = 1024 -> 4 per thread.
  for (int c = tid; c < (D_MODEL / 2) * (NT / 4); c += 256) {
    const int kk = c >> 3;           // pair row (0..127)
    const int j  = (c & 7) << 2;     // col group (0,4,...,28)
    v4f r0 = *(const v4f*)(W + (long)(2 * kk)     * N + n0 + j);
    v4f r1 = *(const v4f*)(W + (long)(2 * kk + 1) * N + n0 + j);
    v4f lo = {r0.x, r1.x, r0.y, r1.y};
    v4f hi = {r0.z, r1.z, r0.w, r1.w};
    *(v4f*)(&Wlds[kk * (NT * 2) + j * 2])     = lo;
    *(v4f*)(&Wlds[kk * (NT * 2) + j * 2 + 4]) = hi;
  }
  __syncthreads();

  const int wave  = tid >> 5;
  const int lane  = tid & 31;
  const int m0    = blockIdx.x * MT + wave * 16;
  const int mrow  = m0 + (lane & 15);
  const int khalf = (lane >> 4) << 1;   // lanes 0-15 -> K,K+1 ; lanes 16-31 -> K+2,K+3
  const int ncol  = lane & 15;

  v8f c0 = {};
  v8f c1 = {};
  const float* Arow = A + (long)mrow * D_MODEL;
  // Per-lane base into the interleaved LDS tile for this lane's K-half/column.
  const float* Bbase = &Wlds[(khalf >> 1) * (NT * 2) + ncol * 2];

  #pragma unroll 8
  for (int k = 0; k < D_MODEL; k += 4) {
    // A fragment: 16x4 f32 (contiguous K pair per lane)
    v2f a = *(const v2f*)(Arow + k + khalf);
    // B fragments: single aligned ds_load_b64 each, no repacking movs
    const float* Bk = Bbase + (k >> 1) * (NT * 2);
    v2f b0 = *(const v2f*)(Bk);
    v2f b1 = *(const v2f*)(Bk + 32);   // next 16-wide column group
    c0 = __builtin_amdgcn_wmma_f32_16x16x4_f32(false, a, false, b0, (short)0, c0, false, false);
    c1 = __builtin_amdgcn_wmma_f32_16x16x4_f32(false, a, false, b1, (short)0, c1, false, false);
  }

  // C layout: lane (0-15) -> column, VGPR r -> row r; lanes 16-31 -> rows r+8.
  const int rbase = m0 + ((lane >> 4) << 3);
  const float bv0 = bias ? bias[n0 + ncol] : 0.0f;
  const float bv1 = bias ? bias[n0 + 16 + ncol] : 0.0f;
  #pragma unroll
  for (int r = 0; r < 8; ++r) {
    C[(long)(rbase + r) * N + n0 + ncol]      = c0[r] + bv0;
    C[(long)(rbase + r) * N + n0 + 16 + ncol] = c1[r] + bv1;
  }
}

// ---------------------------------------------------------------------------
// In-place softmax over L*P = 12 values per (b,q,h). One thread per row-head.
// aw layout: [B*Q, 96] with inner layout [H=8][12].
// ---------------------------------------------------------------------------
__global__ __launch_bounds__(256)
void dta_softmax12(float* __restrict__ aw, int total /* = B*Q*H */) {
  const int t = blockIdx.x * blockDim.x + threadIdx.x;
  if (t >= total) return;
  const int row = t >> 3;
  const int h   = t & 7;
  float* p = aw + (long)row * 96 + h * 12;

  float v[12];
  float mx = -3.402823466e+38f;
  #pragma unroll
  for (int i = 0; i < 12; ++i) { v[i] = p[i]; mx = fmaxf(mx, v[i]); }
  float s = 0.0f;
  #pragma unroll
  for (int i = 0; i < 12; ++i) { v[i] = __expf(v[i] - mx); s += v[i]; }
  const float inv = 1.0f / s;
  #pragma unroll
  for (int i = 0; i < 12; ++i) p[i] = v[i] * inv;
}

// ---------------------------------------------------------------------------
// Deformable sampling + attention accumulation.
//   One wave per query row; lane == HD channel (HD == 32 == wave width).
//   Faithful quirk: the head axis of v_proj is indexed by the POINT index p.
//   out_mid[row, h*32 + hd] = sum_{l,p} aw * lerp(v[b, t, p, hd])
// ---------------------------------------------------------------------------
__global__ __launch_bounds__(256)
void dta_sample(const float* __restrict__ refp, const float* __restrict__ off_raw,
                const float* __restrict__ aw, const float* __restrict__ v_proj,
                float* __restrict__ out_mid) {
  const int wave = (blockIdx.x * blockDim.x + threadIdx.x) >> 5; // query row 0..16383
  const int lane = threadIdx.x & 31;                             // hd channel
  const int b = wave >> 11;      // / 2048
  const int row = wave;          // b*2048 + q

  const float rp = refp[row];
  const int   Ts[3]   = {2048, 1024, 512};
  const long  base[3] = {0, 8L * 2048, 8L * 2048 + 8L * 1024};

  const float* offr = off_raw + (long)row * 96;
  const float* awr  = aw + (long)row * 96;

  #pragma unroll
  for (int h = 0; h < 8; ++h) {
    float acc = 0.0f;
    #pragma unroll
    for (int l = 0; l < 3; ++l) {
      const float Tf = (float)Ts[l];
      const long  rb = base[l] + (long)b * Ts[l];
      #pragma unroll
      for (int p = 0; p < 4; ++p) {
        const int idx = h * 12 + l * 4 + p;
        const float o = offr[idx];
        const float w = awr[idx];
        float pos  = fminf(fmaxf(rp + o / Tf, 0.0f), 1.0f);
        float sidx = pos * (Tf - 1.0f);
        int ifl = (int)sidx;                       // sidx >= 0, trunc == floor
        ifl = min(max(ifl, 0), Ts[l] - 2);
        const float wc = sidx - (float)ifl;
        const long r0 = (rb + ifl) * 256 + p * 32 + lane; // head axis <- p
        const float vf = v_proj[r0];
        const float vc = v_proj[r0 + 256];
        acc += w * ((1.0f - wc) * vf + wc * vc);
      }
    }
    out_mid[(long)row * 256 + h * 32 + lane] = acc;
  }
}

// ---------------------------------------------------------------------------
// Host-side launcher
// ---------------------------------------------------------------------------
extern "C" void kernel_launch(void* const* d_in, const int* in_sizes, int n_in,
                              void* d_out, int out_size, void* d_ws, size_t ws_size,
                              hipStream_t stream) {
  (void)in_sizes; (void)n_in; (void)out_size; (void)ws_size;

  const float* query = (const float*)d_in[0];   // [8,2048,256]
  const float* refp  = (const float*)d_in[1];   // [8,2048]
  const float* val0  = (const float*)d_in[2];   // [8,2048,256]
  const float* val1  = (const float*)d_in[3];   // [8,1024,256]
  const float* val2  = (const float*)d_in[4];   // [8, 512,256]
  const float* Woff  = (const float*)d_in[5];   // [256,96]
  const float* boff  = (const float*)d_in[6];   // [96]
  const float* Waw   = (const float*)d_in[7];   // [256,96]
  const float* baw   = (const float*)d_in[8];   // [96]
  const float* Wv    = (const float*)d_in[9];   // [256,256]
  const float* bv    = (const float*)d_in[10];  // [256]
  const float* Wo    = (const float*)d_in[11];  // [256,256]
  const float* bo    = (const float*)d_in[12];  // [256]
  float*       out   = (float*)d_out;           // [8,2048,256]

  const int BQ = 8 * 2048;                      // 16384 rows

  // Workspace carve-up (all 256B aligned)
  char* ws = (char*)d_ws;
  float* off_raw = (float*)(ws);                                  //  6.29 MB
  float* aw_buf  = (float*)(ws + (size_t)BQ * 96 * 4);            //  6.29 MB
  float* v_proj  = (float*)(ws + (size_t)BQ * 96 * 4 * 2);        // 29.36 MB
  float* out_mid = (float*)(ws + (size_t)BQ * 96 * 4 * 2
                               + (size_t)(BQ + 8192 + 4096) * 256 * 4); // 16.78 MB

  dim3 blk(256);

  // 1) sampling-offset projection: [16384,256] @ [256,96]
  dta_gemm_f32_wmma<<<dim3(BQ / MT, 96 / NT), blk, 0, stream>>>(query, Woff, boff, off_raw, BQ, 96);
  // 2) attention-weight projection
  dta_gemm_f32_wmma<<<dim3(BQ / MT, 96 / NT), blk, 0, stream>>>(query, Waw, baw, aw_buf, BQ, 96);
  // 3) softmax over 12 per (b,q,h)
  dta_softmax12<<<dim3((BQ * 8 + 255) / 256), blk, 0, stream>>>(aw_buf, BQ * 8);
  // 4-6) value projections per level: [B*T,256] @ [256,256]
  dta_gemm_f32_wmma<<<dim3(16384 / MT, 256 / NT), blk, 0, stream>>>(val0, Wv, bv, v_proj, 16384, 256);
  dta_gemm_f32_wmma<<<dim3( 8192 / MT, 256 / NT), blk, 0, stream>>>(val1, Wv, bv, v_proj + (size_t)16384 * 256,  8192, 256);
  dta_gemm_f32_wmma<<<dim3( 4096 / MT, 256 / NT), blk, 0, stream>>>(val2, Wv, bv, v_proj + (size_t)(16384 + 8192) * 256, 4096, 256);
  // 7) gather + lerp + attention-weighted accumulation (one wave per query)
  dta_sample<<<dim3(BQ / 8), blk, 0, stream>>>(refp, off_raw, aw_buf, v_proj, out_mid);
  // 8) output projection: [16384,256] @ [256,256] -> d_out
  dta_gemm_f32_wmma<<<dim3(BQ / MT, 256 / NT), blk, 0, stream>>>(out_mid, Wo, bo, out, BQ, 256);
}